// MyDeConv1D_80367428043218
// MI455X (gfx1250) — compile-verified
//
#include <hip/hip_runtime.h>

// ---------------------------------------------------------------------------
// MyDeConv1D on gfx1250: dilated transposed conv == causal dilated conv.
//   out[b,tau,f] = sum_k sum_c x[b, tau+28-4k, c] * W[c,k,f] + cnt*256*bias[f]
// fp32 end-to-end via V_WMMA_F32_16X16X4_F32 (the only f32 WMMA shape).
// Block = 16 timesteps x 256 filters; 4 wave32s, 4 f-tiles per wave so each
// LDS A-load feeds 4 WMMAs.
// ---------------------------------------------------------------------------

typedef float v2f __attribute__((ext_vector_type(2)));
typedef float v8f __attribute__((ext_vector_type(8)));

#define B_    16
#define T_    2048
#define C_    256
#define K_    8
#define F_    256
#define DIL_  4
#define HALO  28          // (K_-1)*DIL_
#define TT    16          // time-tile (WMMA M)
#define XROWS 44          // TT + HALO rows of x staged per block
#define PITCH 260         // 256 + 4 pad floats -> bank-conflict-free ds_load_b64
#define NTHREADS 128      // 4 wave32s

// --- prepass: W[c,k,f] -> Wt[k,f,c] so B-operand loads are contiguous b64 ---
__global__ __launch_bounds__(256)
void wt_transpose_kernel(const float* __restrict__ w, float* __restrict__ wt) {
    int idx = blockIdx.x * 256 + threadIdx.x;        // linear over [k][f][c]
    if (idx >= K_ * F_ * C_) return;
    int c = idx % C_;
    int f = (idx / C_) % F_;
    int k = idx / (C_ * F_);
    wt[idx] = w[(c * K_ + k) * F_ + f];
}

// --- main kernel ---
__global__ __launch_bounds__(NTHREADS)
void MyDeConv1D_80367428043218_kernel(const float* __restrict__ x,
                                      const float* __restrict__ wt,
                                      const float* __restrict__ bias,
                                      float* __restrict__ out) {
    __shared__ __align__(16) float xs[XROWS * PITCH];

    const int tile = blockIdx.x & (T_ / TT - 1);     // 0..127
    const int b    = blockIdx.x >> 7;                // 0..15
    const int t_lo = tile * TT;                      // first x row needed (p0-28)
    const int tid  = threadIdx.x;

    // Stage x[t_lo .. t_lo+43][0..255] into LDS; zero-fill rows past T_.
    const float4* xg = (const float4*)(x + (size_t)b * T_ * C_);
    for (int i = tid; i < XROWS * (C_ / 4); i += NTHREADS) {
        int row = i >> 6;                            // /64 quads per row
        int cq  = i & 63;
        int t   = t_lo + row;
        float4 v = make_float4(0.f, 0.f, 0.f, 0.f);
        if (t < T_) v = xg[t * (C_ / 4) + cq];
        *(float4*)&xs[row * PITCH + cq * 4] = v;
    }
    __syncthreads();

    const int lane  = tid & 31;
    const int wv    = tid >> 5;                      // wave id 0..3
    const int n16   = lane & 15;                     // N column / M row id
    const int half  = lane >> 4;                     // 0: K={0,1}, 1: K={2,3}
    const int chalf = half * 2;
    const int f0    = wv * 64;                       // this wave's filter base

    v8f acc0 = {};                                   // f0      .. f0+15
    v8f acc1 = {};                                   // f0 + 16 .. f0+31
    v8f acc2 = {};                                   // f0 + 32 .. f0+47
    v8f acc3 = {};                                   // f0 + 48 .. f0+63

    for (int k = 0; k < K_; ++k) {
        // A row for lane (M = n16): x row = (p0 - 4k + m) - t_lo = 28 - 4k + m
        const float* aP  = &xs[(HALO - DIL_ * k + n16) * PITCH + chalf];
        const float* b0P = wt + (((size_t)k * F_ + f0 + n16) * C_ + chalf);
#pragma unroll 4
        for (int cc = 0; cc < C_ / 4; ++cc) {
            v2f a  = *(const v2f*)(aP  + cc * 4);              // ds_load_b64
            v2f b0 = *(const v2f*)(b0P + cc * 4);              // global_load_b64
            v2f b1 = *(const v2f*)(b0P + 16 * C_ + cc * 4);    //   (Wt is L2-resident)
            v2f b2 = *(const v2f*)(b0P + 32 * C_ + cc * 4);
            v2f b3 = *(const v2f*)(b0P + 48 * C_ + cc * 4);
            acc0 = __builtin_amdgcn_wmma_f32_16x16x4_f32(
                       false, a, false, b0, (short)0, acc0, false, false);
            acc1 = __builtin_amdgcn_wmma_f32_16x16x4_f32(
                       false, a, false, b1, (short)0, acc1, false, false);
            acc2 = __builtin_amdgcn_wmma_f32_16x16x4_f32(
                       false, a, false, b2, (short)0, acc2, false, false);
            acc3 = __builtin_amdgcn_wmma_f32_16x16x4_f32(
                       false, a, false, b3, (short)0, acc3, false, false);
        }
    }

    // Epilogue: tap-count-scaled bias (bias added before channel-reduce in the
    // reference => factor C_ per contributing tap), then store.
    const float bias0 = bias[f0 + n16]      * (float)C_;
    const float bias1 = bias[f0 + 16 + n16] * (float)C_;
    const float bias2 = bias[f0 + 32 + n16] * (float)C_;
    const float bias3 = bias[f0 + 48 + n16] * (float)C_;
    float* outb = out + (size_t)b * T_ * F_;
#pragma unroll
    for (int i = 0; i < 8; ++i) {
        int m   = i + half * 8;                      // C/D layout: VGPR i -> M=i / i+8
        int tau = t_lo + m;                          // output timestep
        int d   = (tau + HALO) - (T_ - 1);           // taps with t >= T_ are invalid
        int kmin = d > 0 ? ((d + 3) >> 2) : 0;
        float cnt = (float)(K_ - kmin);
        float* o = outb + (size_t)tau * F_ + f0 + n16;
        o[0]  = acc0[i] + cnt * bias0;
        o[16] = acc1[i] + cnt * bias1;
        o[32] = acc2[i] + cnt * bias2;
        o[48] = acc3[i] + cnt * bias3;
    }
}

extern "C" void kernel_launch(void* const* d_in, const int* in_sizes, int n_in,
                              void* d_out, int out_size, void* d_ws, size_t ws_size,
                              hipStream_t stream) {
    const float* x    = (const float*)d_in[0];   // (B,T,C)
    const float* w    = (const float*)d_in[1];   // (C,K,F)
    const float* bias = (const float*)d_in[2];   // (F,)
    float* out = (float*)d_out;                  // (B,T,F)
    float* wt  = (float*)d_ws;                   // K*F*C floats = 2 MB scratch

    const int nW = K_ * F_ * C_;
    wt_transpose_kernel<<<(nW + 255) / 256, 256, 0, stream>>>(w, wt);

    MyDeConv1D_80367428043218_kernel<<<B_ * (T_ / TT), NTHREADS, 0, stream>>>(
        x, wt, bias, out);
}